// LP_6133213299519
// MI455X (gfx1250) — compile-verified
//
#include <hip/hip_runtime.h>
#include <stdint.h>

#define NCLS 64
#define ALPHAF 0.9f
#define RESWF  0.1f
#define NLAYERS 10

// ---------------------------------------------------------------------------
// Init: deg = 0, winner = -1
// ---------------------------------------------------------------------------
__global__ void init_n(int* __restrict__ deg, int* __restrict__ winner, int n) {
  int i = blockIdx.x * 256 + threadIdx.x;
  if (i < n) { deg[i] = 0; winner[i] = -1; }
}

// ---------------------------------------------------------------------------
// In-degree histogram over destination (col) nodes
// ---------------------------------------------------------------------------
__global__ void deg_count(const int* __restrict__ col, int* __restrict__ deg, int e) {
  int i = blockIdx.x * 256 + threadIdx.x;
  if (i < e) atomicAdd(&deg[col[i]], 1);
}

// ---------------------------------------------------------------------------
// dinv = deg > 0 ? rsqrt(deg) : 0
// ---------------------------------------------------------------------------
__global__ void dinv_kernel(const int* __restrict__ deg, float* __restrict__ dinv, int n) {
  int i = blockIdx.x * 256 + threadIdx.x;
  if (i < n) {
    int d = deg[i];
    dinv[i] = (d > 0) ? rsqrtf((float)d) : 0.0f;
  }
}

// ---------------------------------------------------------------------------
// Exclusive scan, stage A: per-256-block scan of deg -> row_ptr (local), block sums
// ---------------------------------------------------------------------------
__global__ void scan_a(const int* __restrict__ deg, int* __restrict__ row_ptr,
                       int* __restrict__ bsum, int n) {
  __shared__ int tmp[256];
  int t = threadIdx.x;
  int i = blockIdx.x * 256 + t;
  int v = (i < n) ? deg[i] : 0;
  tmp[t] = v;
  __syncthreads();
  #pragma unroll
  for (int off = 1; off < 256; off <<= 1) {
    int x = (t >= off) ? tmp[t - off] : 0;
    __syncthreads();
    tmp[t] += x;
    __syncthreads();
  }
  int incl = tmp[t];
  if (i < n) row_ptr[i] = incl - v;          // local exclusive
  if (t == 255) bsum[blockIdx.x] = incl;     // block total
}

// ---------------------------------------------------------------------------
// Exclusive scan, stage B: single-block scan of block sums (nb <= 1024)
// ---------------------------------------------------------------------------
__global__ void scan_b(const int* __restrict__ bsum, int* __restrict__ boff, int nb) {
  __shared__ int tmp[1024];
  int t = threadIdx.x;
  int v = (t < nb) ? bsum[t] : 0;
  tmp[t] = v;
  __syncthreads();
  #pragma unroll
  for (int off = 1; off < 1024; off <<= 1) {
    int x = (t >= off) ? tmp[t - off] : 0;
    __syncthreads();
    tmp[t] += x;
    __syncthreads();
  }
  if (t < nb) boff[t] = tmp[t] - v;          // exclusive block offsets
}

// ---------------------------------------------------------------------------
// Exclusive scan, stage C: add block offsets, init cursor, write row_ptr[N]
// ---------------------------------------------------------------------------
__global__ void scan_c(int* __restrict__ row_ptr, const int* __restrict__ boff,
                       int* __restrict__ cursor, int n, int e_total) {
  int i = blockIdx.x * 256 + threadIdx.x;
  if (i < n) {
    int v = row_ptr[i] + boff[blockIdx.x];
    row_ptr[i] = v;
    cursor[i]  = v;
  }
  if (i == 0) row_ptr[n] = e_total;
}

// ---------------------------------------------------------------------------
// CSR scatter, grouped by destination node. Edge record = (src, norm) packed
// as int2 (norm is layer-invariant: dinv[src]*dinv[dst]), so the propagation
// loop never touches dinv again.
// ---------------------------------------------------------------------------
__global__ void edge_scatter(const int* __restrict__ row, const int* __restrict__ col,
                             const float* __restrict__ dinv,
                             int* __restrict__ cursor, int2* __restrict__ ew, int e) {
  int i = blockIdx.x * 256 + threadIdx.x;
  if (i < e) {
    int s = row[i];
    int d = col[i];
    int pos = atomicAdd(&cursor[d], 1);
    float w = dinv[s] * dinv[d];
    ew[pos] = make_int2(s, __float_as_int(w));
  }
}

// ---------------------------------------------------------------------------
// Softmax over C=64 classes: one wave32 per row, float2 per lane.
// Also writes res = (1-alpha)*z (fused, saves a full N*C pass).
// ---------------------------------------------------------------------------
__global__ void softmax_rows(const float* __restrict__ y, float* __restrict__ z,
                             float* __restrict__ res, int n) {
  const int wv = threadIdx.x >> 5;
  const int ln = threadIdx.x & 31;
  const int r  = blockIdx.x * 8 + wv;
  if (r >= n) return;
  const float2 v = *(const float2*)(y + (size_t)r * NCLS + ln * 2);
  float m = fmaxf(v.x, v.y);
  #pragma unroll
  for (int o = 16; o > 0; o >>= 1) m = fmaxf(m, __shfl_xor(m, o, 32));
  float e0 = expf(v.x - m);
  float e1 = expf(v.y - m);
  float s = e0 + e1;
  #pragma unroll
  for (int o = 16; o > 0; o >>= 1) s += __shfl_xor(s, o, 32);
  float inv = 1.0f / s;
  e0 *= inv; e1 *= inv;
  *(float2*)(z   + (size_t)r * NCLS + ln * 2) = make_float2(e0, e1);
  *(float2*)(res + (size_t)r * NCLS + ln * 2) = make_float2(RESWF * e0, RESWF * e1);
}

// ---------------------------------------------------------------------------
// Deterministic duplicate resolution: last label index wins
// ---------------------------------------------------------------------------
__global__ void label_winner(const int* __restrict__ mask, int* __restrict__ winner, int l) {
  int i = blockIdx.x * 256 + threadIdx.x;
  if (i < l) atomicMax(&winner[mask[i]], i);
}

// Clamp labeled rows of z to one-hot; keep res consistent (res = 0.1*onehot)
__global__ void label_clamp(const int* __restrict__ mask, const int* __restrict__ ytrue,
                            const int* __restrict__ winner, float* __restrict__ z,
                            float* __restrict__ res, int l) {
  int t = blockIdx.x * 256 + threadIdx.x;
  int i = t >> 6;
  int c = t & 63;
  if (i < l) {
    int r = mask[i];
    if (winner[r] == i) {
      float oh = (c == ytrue[i]) ? 1.0f : 0.0f;
      z  [(size_t)r * NCLS + c] = oh;
      res[(size_t)r * NCLS + c] = RESWF * oh;
    }
  }
}

// ---------------------------------------------------------------------------
// Pull-form SpMM layer: one wave32 per destination node, 2 classes per lane.
// (src, norm) edge records are double-buffered into LDS with the CDNA5
// async-to-LDS path (global_load_async_to_lds_b64 / ASYNCcnt): the DMA for
// chunk k+1 is in flight while chunk k's 32 gather-FMA iterations execute.
// Inner loop: 1 ds b64 broadcast + 1 global b64 gather + 2 FMAs per edge.
// No atomics, no dependent dinv load. out = clip(alpha*agg + res, 0, 1) fused.
// ---------------------------------------------------------------------------
__global__ void lp_pull(const int* __restrict__ row_ptr,
                        const int2* __restrict__ ew,
                        const float* __restrict__ cur,
                        const float* __restrict__ res,
                        float* __restrict__ out, int n) {
  __shared__ int2 sh[8][2][32];
  const int wv = threadIdx.x >> 5;
  const int ln = threadIdx.x & 31;
  const int node = blockIdx.x * 8 + wv;
  if (node >= n) return;                      // uniform per wave

  const int beg = row_ptr[node];
  const int end = row_ptr[node + 1];
  float a0 = 0.0f, a1 = 0.0f;

  int base = beg, buf = 0;
  if (base < end) {
    int cnt = min(32, end - base);
    if (ln < cnt) {
      unsigned ldsa = (unsigned)(size_t)&sh[wv][0][ln];
      asm volatile("global_load_async_to_lds_b64 %0, %1, off"
                   :: "v"(ldsa),
                      "v"((unsigned long long)(size_t)(ew + base + ln))
                   : "memory");
    }
  }
  while (base < end) {
    int cnt = min(32, end - base);
    int nbase = base + cnt;
    asm volatile("s_wait_asynccnt 0" ::: "memory");   // current chunk landed
    if (nbase < end) {                                // prefetch next chunk
      int ncnt = min(32, end - nbase);
      if (ln < ncnt) {
        unsigned ldsa = (unsigned)(size_t)&sh[wv][buf ^ 1][ln];
        asm volatile("global_load_async_to_lds_b64 %0, %1, off"
                     :: "v"(ldsa),
                        "v"((unsigned long long)(size_t)(ew + nbase + ln))
                     : "memory");
      }
    }
    #pragma unroll 4
    for (int it = 0; it < cnt; ++it) {
      const int2 sw = sh[wv][buf][it];                 // (src, norm) broadcast
      const float w = __int_as_float(sw.y);
      const float2 v = *(const float2*)(cur + (size_t)sw.x * NCLS + ln * 2);
      a0 = fmaf(w, v.x, a0);
      a1 = fmaf(w, v.y, a1);
    }
    buf ^= 1;
    base = nbase;
  }

  const float2 r = *(const float2*)(res + (size_t)node * NCLS + ln * 2);
  a0 = fminf(fmaxf(fmaf(ALPHAF, a0, r.x), 0.0f), 1.0f);
  a1 = fminf(fmaxf(fmaf(ALPHAF, a1, r.y), 0.0f), 1.0f);
  *(float2*)(out + (size_t)node * NCLS + ln * 2) = make_float2(a0, a1);
}

// ---------------------------------------------------------------------------
// Host-side launcher
// ---------------------------------------------------------------------------
extern "C" void kernel_launch(void* const* d_in, const int* in_sizes, int n_in,
                              void* d_out, int out_size, void* d_ws, size_t ws_size,
                              hipStream_t stream) {
  const float* y_soft = (const float*)d_in[0];
  const int*   y_true = (const int*)d_in[1];
  const int*   mask   = (const int*)d_in[2];
  const int*   edge   = (const int*)d_in[3];

  const int N = in_sizes[0] / NCLS;
  const int L = in_sizes[1];
  const int E = in_sizes[3] / 2;
  const int* row = edge;       // source
  const int* col = edge + E;   // target

  // ---- carve workspace (256B aligned) ----
  char* p = (char*)d_ws;
  auto carve = [&](size_t bytes) -> void* {
    void* q = (void*)p;
    p += (bytes + 255) & ~(size_t)255;
    return q;
  };
  float* res     = (float*)carve((size_t)N * NCLS * 4);
  float* bufA    = (float*)carve((size_t)N * NCLS * 4);
  float* bufB    = (float*)carve((size_t)N * NCLS * 4);
  float* dinv    = (float*)carve((size_t)N * 4);
  int*   deg     = (int*)  carve((size_t)N * 4);
  int*   winner  = (int*)  carve((size_t)N * 4);
  int*   row_ptr = (int*)  carve((size_t)(N + 1) * 4);
  int*   cursor  = (int*)  carve((size_t)N * 4);
  const int nb   = (N + 255) / 256;
  int*   bsum    = (int*)  carve((size_t)nb * 4);
  int*   boff    = (int*)  carve((size_t)nb * 4);
  int2*  ew      = (int2*) carve((size_t)E * 8);
  (void)ws_size; (void)n_in; (void)out_size;

  const int gN  = (N + 255) / 256;
  const int gE  = (E + 255) / 256;
  const int gW  = (N + 7) / 8;            // one wave32 per node, 8 waves/block
  const int gL  = (L + 255) / 256;
  const int gLC = (L * NCLS + 255) / 256;

  // ---- graph preprocessing (CSR by destination, weights folded in) ----
  init_n      <<<gN, 256, 0, stream>>>(deg, winner, N);
  deg_count   <<<gE, 256, 0, stream>>>(col, deg, E);
  dinv_kernel <<<gN, 256, 0, stream>>>(deg, dinv, N);
  scan_a      <<<nb, 256, 0, stream>>>(deg, row_ptr, bsum, N);
  scan_b      <<<1, 1024, 0, stream>>>(bsum, boff, nb);
  scan_c      <<<nb, 256, 0, stream>>>(row_ptr, boff, cursor, N, E);
  edge_scatter<<<gE, 256, 0, stream>>>(row, col, dinv, cursor, ew, E);

  // ---- smooth(): softmax (+res) + labeled-node clamp (+res) ----
  softmax_rows<<<gW, 256, 0, stream>>>(y_soft, bufA, res, N);
  label_winner<<<gL, 256, 0, stream>>>(mask, winner, L);
  label_clamp <<<gLC, 256, 0, stream>>>(mask, y_true, winner, bufA, res, L);

  // ---- 10 propagation layers, ping-pong; last layer writes d_out ----
  const float* curp = bufA;
  for (int l = 0; l < NLAYERS; ++l) {
    float* outp = (l == NLAYERS - 1) ? (float*)d_out
                                     : ((curp == bufA) ? bufB : bufA);
    lp_pull<<<gW, 256, 0, stream>>>(row_ptr, ew, curp, res, outp, N);
    curp = outp;
  }
}